// MistralSparseMLP_10084583211272
// MI455X (gfx1250) — compile-verified
//
#include <hip/hip_runtime.h>
#include <math.h>

typedef __attribute__((ext_vector_type(2))) float v2f;
typedef __attribute__((ext_vector_type(4))) float v4f;
typedef __attribute__((ext_vector_type(8))) float v8f;

#define H_DIM 4096
#define I_DIM 14336
#define R_DIM 128
#define K_SEL 4096
#define B_TOK 32
#define KC    32           // K-chunk staged in LDS per iteration
#define LDP   (KC + 4)     // padded row: 36 floats -> conflict-free 64-bank access

// async global->LDS DMA: per-lane 16B, tracked by ASYNCcnt
#define ASYNC_B128(ldsoff, gptr)                                            \
  asm volatile("global_load_async_to_lds_b128 %0, %1, off"                  \
               :: "v"(ldsoff), "v"((unsigned long long)(uintptr_t)(gptr))   \
               : "memory")
#define WAIT_ASYNC_8() asm volatile("s_wait_asynccnt 0x8" ::: "memory")
#define WAIT_ASYNC_6() asm volatile("s_wait_asynccnt 0x6" ::: "memory")
#define WAIT_ASYNC_0() asm volatile("s_wait_asynccnt 0x0" ::: "memory")

__device__ __forceinline__ unsigned lds_off32(const void* p) {
  return (unsigned)(uintptr_t)p;   // flat LDS addr low 32 bits = LDS offset
}

// ---------------------------------------------------------------------------
// monotonic key: orders floats as unsigned ints
__device__ __forceinline__ unsigned fkey(float f) {
  unsigned u = __float_as_uint(f);
  return (u & 0x80000000u) ? ~u : (u | 0x80000000u);
}

// ---------------------------------------------------------------------------
// K0: u0[r] = dot(x[0,:], Wr[r,:])   (one block per r)
__global__ __launch_bounds__(256)
void router_u(const float* __restrict__ x, const float* __restrict__ Wr,
              float* __restrict__ u0) {
  __shared__ float red[256];
  const int r = blockIdx.x;
  const float* wrow = Wr + (size_t)r * H_DIM;
  float s = 0.0f;
  for (int h = threadIdx.x; h < H_DIM; h += 256) s += x[h] * wrow[h];
  red[threadIdx.x] = s;
  __syncthreads();
  for (int off = 128; off > 0; off >>= 1) {
    if (threadIdx.x < off) red[threadIdx.x] += red[threadIdx.x + off];
    __syncthreads();
  }
  if (threadIdx.x == 0) u0[r] = red[0];
}

// ---------------------------------------------------------------------------
// K1: probs0[i] = relu(dot(Wsv[i,:], u0) + bsv[i]); one wave32 per row
__global__ __launch_bounds__(256)
void router_probs(const float* __restrict__ Wsv, const float* __restrict__ bsv,
                  const float* __restrict__ u0, float* __restrict__ probs0) {
  const int wave = threadIdx.x >> 5;
  const int lane = threadIdx.x & 31;
  const int i = blockIdx.x * 8 + wave;
  const float* row = Wsv + (size_t)i * R_DIM;
  v4f w = *(const v4f*)&row[lane * 4];
  v4f u = *(const v4f*)&u0[lane * 4];
  float s = w.x * u.x + w.y * u.y + w.z * u.z + w.w * u.w;
  #pragma unroll
  for (int off = 16; off > 0; off >>= 1) s += __shfl_down(s, off, 32);
  if (lane == 0) probs0[i] = fmaxf(s + bsv[i], 0.0f);
}

// ---------------------------------------------------------------------------
// K2: radix bisection for the K-th largest threshold + ordered compaction
__global__ __launch_bounds__(1024)
void topk_select(const float* __restrict__ probs0, int* __restrict__ idxout) {
  __shared__ unsigned cnt;
  __shared__ int scan[1024];
  __shared__ int total_s;
  const int t = threadIdx.x;

  unsigned T = 0;
  for (int bit = 31; bit >= 0; --bit) {
    unsigned cand = T | (1u << bit);
    if (t == 0) cnt = 0u;
    __syncthreads();
    unsigned c = 0;
    for (int i = t; i < I_DIM; i += 1024)
      c += (fkey(probs0[i]) >= cand) ? 1u : 0u;
    atomicAdd(&cnt, c);
    __syncthreads();
    if (cnt >= (unsigned)K_SEL) T = cand;   // uniform decision across block
    __syncthreads();
  }

  // ordered compaction: contiguous segment of 14 elems per thread + block scan
  const int SEG = I_DIM / 1024;             // 14
  const int base = t * SEG;
  int cme = 0;
  for (int j = 0; j < SEG; ++j)
    if (fkey(probs0[base + j]) >= T) ++cme;
  scan[t] = cme;
  __syncthreads();
  for (int off = 1; off < 1024; off <<= 1) {
    int v = (t >= off) ? scan[t - off] : 0;
    __syncthreads();
    scan[t] += v;
    __syncthreads();
  }
  if (t == 1023) total_s = scan[1023];
  int pos = scan[t] - cme;                   // exclusive prefix
  for (int j = 0; j < SEG; ++j) {
    int i = base + j;
    if (fkey(probs0[i]) >= T) {
      if (pos < K_SEL) idxout[pos] = i;
      ++pos;
    }
  }
  __syncthreads();
  for (int p = total_s + t; p < K_SEL; p += 1024) idxout[p] = 0; // safety fill
}

// ---------------------------------------------------------------------------
// K3: sparse gate/up SwiGLU.  Block = 64 threads (2 waves), tile = [32 x 16].
// Double-buffered LDS fed by async global->LDS DMA, f32 WMMA over K=H.
__global__ __launch_bounds__(64)
void gateup_wmma(const float* __restrict__ x, const float* __restrict__ Wg,
                 const float* __restrict__ Wu, const int* __restrict__ idx,
                 float* __restrict__ mu) {
  __shared__ float xs[2][B_TOK][LDP];
  __shared__ float gs[2][16][LDP];
  __shared__ float us[2][16][LDP];
  const int t = threadIdx.x;
  const int lane = t & 31;
  const int m_tile = t >> 5;                 // 0..1
  const int nbase = blockIdx.x * 16;

  const int xr = t >> 1;                     // 0..31 token row
  const int xc = (t & 1) << 4;               // 0 or 16
  const int wr = t >> 2;                     // 0..15 neuron row
  const int wc = (t & 3) << 2;               // 0,4,8,12

  int nrow = idx[nbase + wr];
  nrow = min(max(nrow, 0), I_DIM - 1);       // clamp against stale/poison
  const float* gp = Wg + (size_t)nrow * H_DIM;
  const float* up = Wu + (size_t)nrow * H_DIM;
  const float* xp = x + (size_t)xr * H_DIM;

  unsigned xo0 = lds_off32(&xs[0][xr][xc]), xo1 = lds_off32(&xs[1][xr][xc]);
  unsigned go0 = lds_off32(&gs[0][wr][wc]), go1 = lds_off32(&gs[1][wr][wc]);
  unsigned uo0 = lds_off32(&us[0][wr][wc]), uo1 = lds_off32(&us[1][wr][wc]);

  v8f accg = {};
  v8f accu = {};

  const int am = m_tile * 16 + (lane & 15);  // A: lane -> M, half -> K pair
  const int bn = lane & 15;                  // B: lane -> N, half -> K pair
  const int ko = (lane >> 4) << 1;

  // prologue: stream chunk 0 into buffer 0
  {
    ASYNC_B128(xo0 +  0, xp + xc + 0);
    ASYNC_B128(xo0 + 16, xp + xc + 4);
    ASYNC_B128(xo0 + 32, xp + xc + 8);
    ASYNC_B128(xo0 + 48, xp + xc + 12);
    ASYNC_B128(go0 +  0, gp + wc);
    ASYNC_B128(go0 + 64, gp + wc + 16);
    ASYNC_B128(uo0 +  0, up + wc);
    ASYNC_B128(uo0 + 64, up + wc + 16);
  }

  const int NCH = H_DIM / KC;                // 128 chunks
  for (int c = 0; c < NCH; ++c) {
    const int b = c & 1;
    __syncthreads();                         // everyone done reading spare buf
    if (c + 1 < NCH) {
      const int h1 = (c + 1) * KC;
      unsigned xo = b ? xo0 : xo1;
      unsigned go = b ? go0 : go1;
      unsigned uo = b ? uo0 : uo1;
      ASYNC_B128(xo +  0, xp + h1 + xc + 0);
      ASYNC_B128(xo + 16, xp + h1 + xc + 4);
      ASYNC_B128(xo + 32, xp + h1 + xc + 8);
      ASYNC_B128(xo + 48, xp + h1 + xc + 12);
      ASYNC_B128(go +  0, gp + h1 + wc);
      ASYNC_B128(go + 64, gp + h1 + wc + 16);
      ASYNC_B128(uo +  0, up + h1 + wc);
      ASYNC_B128(uo + 64, up + h1 + wc + 16);
      WAIT_ASYNC_8();                        // chunk c landed (in-order)
    } else {
      WAIT_ASYNC_0();
    }
    __syncthreads();                         // chunk c visible to both waves
    #pragma unroll
    for (int k = 0; k < KC; k += 4) {
      v2f a  = *(const v2f*)&xs[b][am][k + ko];
      v2f bg = *(const v2f*)&gs[b][bn][k + ko];
      v2f bu = *(const v2f*)&us[b][bn][k + ko];
      accg = __builtin_amdgcn_wmma_f32_16x16x4_f32(false, a, false, bg,
                                                   (short)0, accg, false, false);
      accu = __builtin_amdgcn_wmma_f32_16x16x4_f32(false, a, false, bu,
                                                   (short)0, accu, false, false);
    }
  }

  // epilogue: silu(g)*u scattered into dense mu[token, neuron]
  int ng = idx[nbase + (lane & 15)];
  ng = min(max(ng, 0), I_DIM - 1);
  const int m0 = m_tile * 16 + ((lane >> 4) << 3);   // C/D: M = v + 8*half
  #pragma unroll
  for (int v = 0; v < 8; ++v) {
    float g = accg[v];
    float s = g / (1.0f + __expf(-g));
    mu[(size_t)(m0 + v) * I_DIM + ng] = s * accu[v];
  }
}

// ---------------------------------------------------------------------------
// K4: dense down projection out = mu @ Wd^T.  Same async double-buffer scheme.
__global__ __launch_bounds__(64)
void down_wmma(const float* __restrict__ mu, const float* __restrict__ Wd,
               float* __restrict__ out) {
  __shared__ float ms[2][B_TOK][LDP];
  __shared__ float ds_[2][16][LDP];
  const int t = threadIdx.x;
  const int lane = t & 31;
  const int m_tile = t >> 5;
  const int hbase = blockIdx.x * 16;

  const int xr = t >> 1;
  const int xc = (t & 1) << 4;
  const int wr = t >> 2;
  const int wc = (t & 3) << 2;

  const float* mp = mu + (size_t)xr * I_DIM;
  const float* dp = Wd + (size_t)(hbase + wr) * I_DIM;

  unsigned mo0 = lds_off32(&ms[0][xr][xc]), mo1 = lds_off32(&ms[1][xr][xc]);
  unsigned do0 = lds_off32(&ds_[0][wr][wc]), do1 = lds_off32(&ds_[1][wr][wc]);

  v8f acc = {};
  const int am = m_tile * 16 + (lane & 15);
  const int bn = lane & 15;
  const int ko = (lane >> 4) << 1;

  {
    ASYNC_B128(mo0 +  0, mp + xc + 0);
    ASYNC_B128(mo0 + 16, mp + xc + 4);
    ASYNC_B128(mo0 + 32, mp + xc + 8);
    ASYNC_B128(mo0 + 48, mp + xc + 12);
    ASYNC_B128(do0 +  0, dp + wc);
    ASYNC_B128(do0 + 64, dp + wc + 16);
  }

  const int NCH = I_DIM / KC;                // 448 chunks
  for (int c = 0; c < NCH; ++c) {
    const int b = c & 1;
    __syncthreads();
    if (c + 1 < NCH) {
      const int i1 = (c + 1) * KC;
      unsigned mo = b ? mo0 : mo1;
      unsigned dox = b ? do0 : do1;
      ASYNC_B128(mo +  0, mp + i1 + xc + 0);
      ASYNC_B128(mo + 16, mp + i1 + xc + 4);
      ASYNC_B128(mo + 32, mp + i1 + xc + 8);
      ASYNC_B128(mo + 48, mp + i1 + xc + 12);
      ASYNC_B128(dox +  0, dp + i1 + wc);
      ASYNC_B128(dox + 64, dp + i1 + wc + 16);
      WAIT_ASYNC_6();
    } else {
      WAIT_ASYNC_0();
    }
    __syncthreads();
    #pragma unroll
    for (int k = 0; k < KC; k += 4) {
      v2f a = *(const v2f*)&ms[b][am][k + ko];
      v2f w = *(const v2f*)&ds_[b][bn][k + ko];
      acc = __builtin_amdgcn_wmma_f32_16x16x4_f32(false, a, false, w,
                                                  (short)0, acc, false, false);
    }
  }

  const int hn = hbase + (lane & 15);
  const int m0 = m_tile * 16 + ((lane >> 4) << 3);
  #pragma unroll
  for (int v = 0; v < 8; ++v)
    out[(size_t)(m0 + v) * H_DIM + hn] = acc[v];
}

// ---------------------------------------------------------------------------
extern "C" void kernel_launch(void* const* d_in, const int* in_sizes, int n_in,
                              void* d_out, int out_size, void* d_ws, size_t ws_size,
                              hipStream_t stream) {
  (void)in_sizes; (void)n_in; (void)out_size; (void)ws_size;
  const float* x   = (const float*)d_in[0];
  const float* Wg  = (const float*)d_in[1];
  const float* Wu  = (const float*)d_in[2];
  const float* Wd  = (const float*)d_in[3];
  const float* Wr  = (const float*)d_in[4];
  const float* Wsv = (const float*)d_in[5];
  const float* bsv = (const float*)d_in[6];
  float* out = (float*)d_out;
  float* ws  = (float*)d_ws;

  float* u0     = ws;                              // 128 f32
  float* probs0 = ws + 128;                        // 14336 f32
  int*   idx    = (int*)(ws + 128 + I_DIM);        // 4096 i32
  float* mu     = ws + 128 + I_DIM + K_SEL;        // 32*14336 f32

  hipMemsetAsync(mu, 0, (size_t)B_TOK * I_DIM * sizeof(float), stream);
  router_u<<<R_DIM, 256, 0, stream>>>(x, Wr, u0);
  router_probs<<<I_DIM / 8, 256, 0, stream>>>(Wsv, bsv, u0, probs0);
  topk_select<<<1, 1024, 0, stream>>>(probs0, idx);
  gateup_wmma<<<K_SEL / 16, 64, 0, stream>>>(x, Wg, Wu, idx, mu);
  down_wmma<<<H_DIM / 16, 64, 0, stream>>>(mu, Wd, out);
}